// HashEncoding_24223615549753
// MI455X (gfx1250) — compile-verified
//
#include <hip/hip_runtime.h>
#include <stdint.h>

// ---------------- configuration (matches reference) ----------------
#define HASH_SIZE   (1u << 20)
#define HASH_MASK   (HASH_SIZE - 1u)
#define NLEV        5
#define NFEAT       4
#define PTS_BLK     256
#define TILE_ELEMS  (PTS_BLK * NLEV * NFEAT)   // 5120 floats = 20 KB per block
#define PI2         19349663u
#define PI3         83492791u
#define PRECOND     10.0f

typedef float        fvec4 __attribute__((ext_vector_type(4)));
typedef unsigned int v4u   __attribute__((ext_vector_type(4)));
typedef int          v4i   __attribute__((ext_vector_type(4)));
typedef int          v8i   __attribute__((ext_vector_type(8)));

#if defined(__gfx1250__) && __has_builtin(__builtin_amdgcn_tensor_store_from_lds)
#define USE_TDM 1
#else
#define USE_TDM 0
#endif

__global__ __launch_bounds__(PTS_BLK)
void HashEncoding_24223615549753_kernel(const float* __restrict__ x,
                                        const float* __restrict__ tables,
                                        float* __restrict__ out,
                                        int n)
{
    __shared__ float tile[TILE_ELEMS];          // [pt][level][4] tile staging, 20 KB

    const int tid = threadIdx.x;
    const long long gi_raw = (long long)blockIdx.x * PTS_BLK + tid;
    const bool valid = gi_raw < (long long)n;
    const long long gi = valid ? gi_raw : (long long)(n - 1);   // clamp for safe loads

    // Streaming, read-once input: non-temporal so it never evicts hash tables from L2.
    const float xin = __builtin_nontemporal_load(x + gi * 3 + 0);
    const float yin = __builtin_nontemporal_load(x + gi * 3 + 1);
    const float zin = __builtin_nontemporal_load(x + gi * 3 + 2);

    // bbox [-2,2] -> [0,1]  (exact: /4 == *0.25)
    const float x01 = (xin + 2.0f) * 0.25f;
    const float y01 = (yin + 2.0f) * 0.25f;
    const float z01 = (zin + 2.0f) * 0.25f;

    const float gsz[NLEV] = {128.0f, 256.0f, 512.0f, 1024.0f, 2048.0f};

    fvec4* tile4 = (fvec4*)tile;

#pragma unroll
    for (int l = 0; l < NLEV; ++l) {
        const float g = gsz[l];
        // locations = x01*g, then -0.5 with half_pixel_center handling
        const float vx = x01 * g - 0.5f;
        const float vy = y01 * g - 0.5f;
        const float vz = z01 * g - 0.5f;
        const float fx = floorf(vx), fy = floorf(vy), fz = floorf(vz);
        const float wx1 = vx - fx, wy1 = vy - fy, wz1 = vz - fz;
        const float wx0 = 1.0f - wx1, wy0 = 1.0f - wy1, wz0 = 1.0f - wz1;

        // uint32 wraparound semantics identical to jnp uint32 arithmetic
        const uint32_t bx = (uint32_t)(int)fx;
        const uint32_t by = (uint32_t)(int)fy;
        const uint32_t bz = (uint32_t)(int)fz;

        // only two u32 multiplies per level; +1 corners by add
        const uint32_t hy0 = by * PI2, hy1 = hy0 + PI2;
        const uint32_t hz0 = bz * PI3, hz1 = hz0 + PI3;
        const uint32_t bx1 = bx + 1u;

        const uint32_t yz00 = hy0 ^ hz0, yz01 = hy0 ^ hz1;
        const uint32_t yz10 = hy1 ^ hz0, yz11 = hy1 ^ hz1;

        const uint32_t i000 = (bx  ^ yz00) & HASH_MASK;
        const uint32_t i001 = (bx  ^ yz01) & HASH_MASK;
        const uint32_t i010 = (bx  ^ yz10) & HASH_MASK;
        const uint32_t i011 = (bx  ^ yz11) & HASH_MASK;
        const uint32_t i100 = (bx1 ^ yz00) & HASH_MASK;
        const uint32_t i101 = (bx1 ^ yz01) & HASH_MASK;
        const uint32_t i110 = (bx1 ^ yz10) & HASH_MASK;
        const uint32_t i111 = (bx1 ^ yz11) & HASH_MASK;

        // Table stays RT (default hint): resident in 192 MB L2 (all 5 tables = 80 MB).
        const fvec4* __restrict__ tbl =
            (const fvec4*)(tables + (size_t)l * HASH_SIZE * NFEAT);

        // Issue all 8 gathers (global_load_b128) before consuming -> max MLP.
        const fvec4 f000 = tbl[i000];
        const fvec4 f001 = tbl[i001];
        const fvec4 f010 = tbl[i010];
        const fvec4 f011 = tbl[i011];
        const fvec4 f100 = tbl[i100];
        const fvec4 f101 = tbl[i101];
        const fvec4 f110 = tbl[i110];
        const fvec4 f111 = tbl[i111];

        const float w00 = wy0 * wz0, w01 = wy0 * wz1;
        const float w10 = wy1 * wz0, w11 = wy1 * wz1;

        // reference corner order: (dx,dy,dz) = 000,001,010,011,100,101,110,111
        fvec4 acc = f000 * (wx0 * w00);
        acc += f001 * (wx0 * w01);
        acc += f010 * (wx0 * w10);
        acc += f011 * (wx0 * w11);
        acc += f100 * (wx1 * w00);
        acc += f101 * (wx1 * w01);
        acc += f110 * (wx1 * w10);
        acc += f111 * (wx1 * w11);

        tile4[tid * NLEV + l] = acc * PRECOND;   // ds_store_b128
    }

    __syncthreads();

    const bool fullTile = ((long long)(blockIdx.x + 1) * PTS_BLK) <= (long long)n;
    float* __restrict__ obase = out + (size_t)blockIdx.x * TILE_ELEMS;

#if USE_TDM
    if (fullTile) {
        if (tid == 0) {
            // Tensor Data Mover: DMA the 20 KB LDS tile -> global as one 1-row 2D tile.
            const uint64_t gaddr  = (uint64_t)(uintptr_t)obase;
            const uint32_t ldsoff = (uint32_t)(uintptr_t)&tile[0]; // low 32b of flat = LDS offset

            v4u g0;
            g0.x = 1u;                                       // count=1 valid descriptor
            g0.y = ldsoff;                                   // lds_addr
            g0.z = (uint32_t)gaddr;                          // global_addr[31:0]
            g0.w = (uint32_t)((gaddr >> 32) & 0x01FFFFFFu)   // global_addr[56:32]
                 | (2u << 30);                               // type=2 (image)

            v8i g1;
            g1[0] = (int)(2u << 16);                           // data_size=2 (4 bytes)
            g1[1] = (int)(((uint32_t)TILE_ELEMS & 0xFFFFu) << 16); // tensor_dim0 lo16
            g1[2] = (int)(((uint32_t)TILE_ELEMS >> 16) | (1u << 16)); // dim0 hi | tensor_dim1=1
            g1[3] = (int)(((uint32_t)TILE_ELEMS & 0xFFFFu) << 16);    // tile_dim0=5120
            g1[4] = 1;                                         // tile_dim1=1, tile_dim2=0
            g1[5] = (int)TILE_ELEMS;                           // tensor_dim0_stride lo32
            g1[6] = 0;
            g1[7] = 0;

            v4i z4 = {0, 0, 0, 0};                             // groups 2/3 unused (2D)
            v8i z8 = {0, 0, 0, 0, 0, 0, 0, 0};                 // unused extra group (clang-23 form)
            __builtin_amdgcn_tensor_store_from_lds(g0, g1, z4, z4, z8, 0);
#if __has_builtin(__builtin_amdgcn_s_wait_tensorcnt)
            __builtin_amdgcn_s_wait_tensorcnt(0);              // drain before LDS dealloc
#endif
        }
    } else {
        // rare ragged tail: guarded scalar stores
        for (int k = tid; k < TILE_ELEMS; k += PTS_BLK) {
            long long pt = (long long)blockIdx.x * PTS_BLK + k / (NLEV * NFEAT);
            if (pt < (long long)n) __builtin_nontemporal_store(tile[k], obase + k);
        }
    }
#else
    if (fullTile) {
        // coalesced 16B/lane non-temporal streaming store of the tile
        const fvec4* t4 = (const fvec4*)tile;
        fvec4* o4 = (fvec4*)obase;
        for (int k = tid; k < TILE_ELEMS / 4; k += PTS_BLK)
            __builtin_nontemporal_store(t4[k], o4 + k);
    } else {
        for (int k = tid; k < TILE_ELEMS; k += PTS_BLK) {
            long long pt = (long long)blockIdx.x * PTS_BLK + k / (NLEV * NFEAT);
            if (pt < (long long)n) __builtin_nontemporal_store(tile[k], obase + k);
        }
    }
#endif
}

extern "C" void kernel_launch(void* const* d_in, const int* in_sizes, int n_in,
                              void* d_out, int out_size, void* d_ws, size_t ws_size,
                              hipStream_t stream)
{
    const float* x      = (const float*)d_in[0];   // [N,3] fp32
    const float* tables = (const float*)d_in[1];   // [5, 2^20, 4] fp32
    float* out          = (float*)d_out;           // [N, 20] fp32
    const int n = in_sizes[0] / 3;
    const int blocks = (n + PTS_BLK - 1) / PTS_BLK;
    HashEncoding_24223615549753_kernel<<<blocks, PTS_BLK, 0, stream>>>(x, tables, out, n);
}